// KNNBuffer_AHA_87144886436102
// MI455X (gfx1250) — compile-verified
//
#include <hip/hip_runtime.h>

// inputs [B,D] f32, buffer [B,M,D] f32.
// out[b,:] = buffer[b, argmax_m ||buffer[b,m]-inputs[b]||, :]
#define B_ 64
#define M_ 2048
#define D_ 512

#define TILE_M 128            // rows per block (8 wave32 x 16 rows)
#define TILE_K 64             // d-chunk staged per pipeline stage
#define NCHUNK (D_ / TILE_K)  // 8
#define SPLITS (M_ / TILE_M)  // 16 m-splits -> grid (64, 16) = 1024 blocks
#define ROWSTRIDE 68          // 64 + 4 dword pad: conflict-free ds reads, 16B aligned

typedef __attribute__((ext_vector_type(2))) float v2f;
typedef __attribute__((ext_vector_type(4))) float f4;
typedef __attribute__((ext_vector_type(8))) float v8f;

// ---------------- Kernel 1: per-(b, m-split) partial argmax via WMMA ----------------
__global__ __launch_bounds__(256)
void knn_partial(const float* __restrict__ inputs,
                 const float* __restrict__ buffer,
                 float* __restrict__ ws)
{
    __shared__ float xs[D_];
    __shared__ float tileZ[TILE_M * ROWSTRIDE];
    __shared__ float redD[8];
    __shared__ int   redI[8];

    const int b    = blockIdx.x;
    const int m0   = blockIdx.y * TILE_M;
    const int t    = threadIdx.x;
    const int lane = t & 31;
    const int w    = t >> 5;
    const int rw   = w * 16;
    const int lrow = lane & 15;
    const int half = lane >> 4;

    // Per-thread staging geometry: slot i covers (row, 4-float column group).
    int rowv[8], c4v[8];
    #pragma unroll
    for (int i = 0; i < 8; ++i) {
        int slot = t + 256 * i;        // 0..2047
        rowv[i] = slot >> 4;           // 16 float4 per 64-wide row
        c4v[i]  = (slot & 15) << 2;
    }
    const float* gbase = buffer + ((size_t)b * M_ + (size_t)m0) * D_;

    // Issue chunk-0 global loads immediately (independent of xs).
    f4 y[8];
    #pragma unroll
    for (int i = 0; i < 8; ++i)
        y[i] = __builtin_nontemporal_load(
                   (const f4*)(gbase + (size_t)rowv[i] * D_ + c4v[i]));

    // Stage x[b,:] (overlaps with the loads above).
    xs[t]       = inputs[b * D_ + t];
    xs[t + 256] = inputs[b * D_ + t + 256];
    __syncthreads();   // xs visible before first subtract

    v8f acc = {};
    const float* zbase = &tileZ[(rw + lrow) * ROWSTRIDE + 2 * half];

    // Software-pipelined: chunk c+1's global loads issue before chunk c's WMMAs,
    // hiding HBM latency behind 16 WMMAs + barriers.
    #pragma unroll
    for (int c = 0; c < NCHUNK; ++c) {
        const int k0 = c * TILE_K;

        f4 z[8];
        #pragma unroll
        for (int i = 0; i < 8; ++i) {   // waits chunk-c loads here
            z[i].x = y[i].x - xs[k0 + c4v[i] + 0];
            z[i].y = y[i].y - xs[k0 + c4v[i] + 1];
            z[i].z = y[i].z - xs[k0 + c4v[i] + 2];
            z[i].w = y[i].w - xs[k0 + c4v[i] + 3];
        }

        if (c > 0) __syncthreads();     // prior chunk's WMMA reads done before overwrite
        #pragma unroll
        for (int i = 0; i < 8; ++i)
            *(f4*)&tileZ[rowv[i] * ROWSTRIDE + c4v[i]] = z[i];   // ds_store_b128

        if (c + 1 < NCHUNK) {           // prefetch next chunk while we compute
            #pragma unroll
            for (int i = 0; i < 8; ++i)
                y[i] = __builtin_nontemporal_load(
                           (const f4*)(gbase + (size_t)rowv[i] * D_
                                       + (k0 + TILE_K) + c4v[i]));
        }
        __syncthreads();                // staging visible

        // C += Z_chunk * Z_chunk^T  (A-frag == B-frag register image for B = A^T).
        #pragma unroll
        for (int kk = 0; kk < TILE_K; kk += 4) {
            v2f a = *(const v2f*)(zbase + kk);   // ds_load_b64, conflict-free
            acc = __builtin_amdgcn_wmma_f32_16x16x4_f32(
                    false, a, false, a, (short)0, acc, false, false);
        }
    }

    // Diagonal of C = squared distances. row j<8 -> lane j, VGPR j;
    // row j>=8 -> lane j+16, VGPR j-8.
    float bd = -__builtin_inff();
    int   bi = 0x7fffffff;
    const bool diagLane = (lane < 8) || (lane >= 24);
    if (diagLane) {
        const int vg = (lane < 8) ? lane : (lane - 24);
        const int dj = (lane < 8) ? lane : (lane - 16);
        bd = acc[vg];
        bi = m0 + rw + dj;
    }

    // Wave argmax butterfly (wave32), tie-break to lower index.
    #pragma unroll
    for (int off = 16; off > 0; off >>= 1) {
        float od = __shfl_xor(bd, off, 32);
        int   oi = __shfl_xor(bi, off, 32);
        if (od > bd || (od == bd && oi < bi)) { bd = od; bi = oi; }
    }
    if (lane == 0) { redD[w] = bd; redI[w] = bi; }
    __syncthreads();
    if (t == 0) {
        float mbd = redD[0]; int mbi = redI[0];
        for (int i = 1; i < 8; ++i) {
            if (redD[i] > mbd || (redD[i] == mbd && redI[i] < mbi)) {
                mbd = redD[i]; mbi = redI[i];
            }
        }
        float* dst = ws + (size_t)(b * SPLITS + blockIdx.y) * 2;
        dst[0] = mbd;
        dst[1] = __int_as_float(mbi);
    }
}

// ---------------- Kernel 2: reduce 16 split candidates per row, gather ----------------
__global__ __launch_bounds__(256)
void knn_reduce_gather(const float* __restrict__ buffer,
                       const float* __restrict__ ws,
                       float* __restrict__ out)
{
    __shared__ int bestSh;
    const int b = blockIdx.x;
    const int t = threadIdx.x;

    if (t == 0) {
        float mbd = -__builtin_inff();
        int   mbi = 0x7fffffff;
        for (int s = 0; s < SPLITS; ++s) {
            const float* e = ws + (size_t)(b * SPLITS + s) * 2;
            float d = e[0];
            int   i = __float_as_int(e[1]);
            if (d > mbd || (d == mbd && i < mbi)) { mbd = d; mbi = i; }
        }
        bestSh = mbi;
    }
    __syncthreads();

    const float* src = buffer + ((size_t)b * M_ + bestSh) * D_;
    out[b * D_ + t]       = src[t];
    out[b * D_ + t + 256] = src[t + 256];
}

extern "C" void kernel_launch(void* const* d_in, const int* in_sizes, int n_in,
                              void* d_out, int out_size, void* d_ws, size_t ws_size,
                              hipStream_t stream)
{
    (void)in_sizes; (void)n_in; (void)out_size; (void)ws_size;
    const float* inputs = (const float*)d_in[0];   // [B, D]
    const float* buffer = (const float*)d_in[1];   // [B, M, D]
    float* out = (float*)d_out;                    // [B, D]
    float* ws  = (float*)d_ws;                     // B*SPLITS*2 floats = 8 KB

    dim3 grid1(B_, SPLITS);
    knn_partial<<<grid1, 256, 0, stream>>>(inputs, buffer, ws);
    knn_reduce_gather<<<B_, 256, 0, stream>>>(buffer, ws, out);
}